// Encoder_target_47270410060158
// MI455X (gfx1250) — compile-verified
//
#include <hip/hip_runtime.h>
#include <hip/hip_bf16.h>

// GCN forward for MI455X (gfx1250, wave32).
//  - 3 dense 50000x128 @ 128x128 fp32 GEMMs via V_WMMA_F32_16X16X4_F32
//  - edge scatter with global_atomic_add_f32 (working set L2-resident: 192MB L2)
//  - bias + self-loop fused into output-init kernel
// d_out layout: [total | x0 | e1 | e2 | e3], each 50000x128 f32.
// The "total" region doubles as the per-layer y scratch until the final sum.

typedef __attribute__((ext_vector_type(2))) float v2f;
typedef __attribute__((ext_vector_type(8))) float v8f;

#define GCN_D 128
#define GCN_N 50000

// ---------------- degree / dinv ----------------
__global__ void k_deg_init(float* __restrict__ deg, int n) {
  int i = blockIdx.x * blockDim.x + threadIdx.x;
  if (i < n) deg[i] = 1.0f;  // self loop contributes 1
}

__global__ void k_deg_edges(const int* __restrict__ dst, float* __restrict__ deg, int E) {
  int e = blockIdx.x * blockDim.x + threadIdx.x;
  if (e < E) atomicAdd(&deg[dst[e]], 1.0f);
}

__global__ void k_dinv(float* __restrict__ deg, int n) {
  int i = blockIdx.x * blockDim.x + threadIdx.x;
  if (i < n) deg[i] = rsqrtf(deg[i]);  // deg >= 1 always (self loops)
}

// ---------------- x0 copy ----------------
__global__ void k_copy4(const float4* __restrict__ src, float4* __restrict__ out, int n4) {
  int i = blockIdx.x * blockDim.x + threadIdx.x;
  if (i < n4) out[i] = src[i];
}

// ---------------- GEMM: Y = X @ W, X: nRows x 128, W: 128 x 128 ----------------
// One wave per 16x16 output tile. K-loop: 32 steps of V_WMMA_F32_16X16X4_F32.
// A (16x4 f32, 2 VGPRs): lanes 0-15 hold {K+0,K+1} for M=lane; lanes 16-31 hold
// {K+2,K+3} for M=lane-16  -> one contiguous float2 load per step.
// B (4x16, 2 VGPRs): v0/v1 = rows K+kOff / K+kOff+1, N = lane&15 (kOff=2 for hi half).
// C/D (16x16, 8 VGPRs): VGPR r -> row 16*tile + r (+8 for lanes 16-31), N = lane&15.
__global__ void k_gemm_wmma(const float* __restrict__ X, const float* __restrict__ W,
                            float* __restrict__ Y, int nRows) {
  const int lane = threadIdx.x & 31;
  const int wave = threadIdx.x >> 5;
  const int tile = blockIdx.x * (blockDim.x >> 5) + wave;
  const int nColTiles = GCN_D / 16;                 // 8
  const int rowTile = tile / nColTiles;
  const int colTile = tile - rowTile * nColTiles;
  if (rowTile * 16 >= nRows) return;                // wave-uniform: EXEC stays all-1s

  const int half = lane >> 4;                       // 0: lanes 0-15, 1: lanes 16-31
  const int l16  = lane & 15;
  const int row  = rowTile * 16 + l16;
  const int col  = colTile * 16 + l16;
  const int kOff = half * 2;

  const float* __restrict__ xrow = X + (size_t)row * GCN_D;
  v8f c = {};
#pragma unroll
  for (int k0 = 0; k0 < GCN_D; k0 += 4) {
    v2f a = *(const v2f*)(xrow + k0 + kOff);        // global_load_b64
    v2f b;
    b.x = W[(size_t)(k0 + kOff)     * GCN_D + col];
    b.y = W[(size_t)(k0 + kOff + 1) * GCN_D + col];
    c = __builtin_amdgcn_wmma_f32_16x16x4_f32(
        /*neg_a=*/false, a, /*neg_b=*/false, b,
        /*c_mod=*/(short)0, c, /*reuse_a=*/false, /*reuse_b=*/false);
  }

  const int mBase = rowTile * 16 + half * 8;
#pragma unroll
  for (int r = 0; r < 8; ++r) {
    Y[(size_t)(mBase + r) * GCN_D + col] = c[r];
  }
}

// ---------------- out = bias + self-loop: O[i] = b + y[i]*dinv[i]^2 ----------------
__global__ void k_self_bias(const float4* __restrict__ y, const float* __restrict__ dinv,
                            const float4* __restrict__ bias, float4* __restrict__ O, int n) {
  int gid = blockIdx.x * blockDim.x + threadIdx.x;
  int i = gid >> 5, c = gid & 31;                   // 32 float4 per row of 128
  if (i >= n) return;
  float w = dinv[i]; w = w * w;
  float4 v = y[(size_t)i * 32 + c];
  float4 b = bias[c];
  float4 o;
  o.x = b.x + v.x * w; o.y = b.y + v.y * w;
  o.z = b.z + v.z * w; o.w = b.w + v.w * w;
  O[(size_t)i * 32 + c] = o;
}

// ---------------- edge scatter: O[dst] += y[src] * dinv[src]*dinv[dst] ----------------
__global__ void k_scatter(const float4* __restrict__ y, const int* __restrict__ src,
                          const int* __restrict__ dst, const float* __restrict__ dinv,
                          float* __restrict__ O, int E) {
  int gid = blockIdx.x * blockDim.x + threadIdx.x;
  int e = gid >> 5, c = gid & 31;
  if (e >= E) return;
  int s = src[e], d = dst[e];
  float w = dinv[s] * dinv[d];
  float4 v = y[(size_t)s * 32 + c];
  float* o = O + (size_t)d * GCN_D + c * 4;
  atomicAdd(o + 0, v.x * w);
  atomicAdd(o + 1, v.y * w);
  atomicAdd(o + 2, v.z * w);
  atomicAdd(o + 3, v.w * w);
}

// ---------------- total = x0 + e1 + e2 + e3 ----------------
__global__ void k_total(const float4* __restrict__ a, const float4* __restrict__ b,
                        const float4* __restrict__ c, const float4* __restrict__ d,
                        float4* __restrict__ out, int n4) {
  int i = blockIdx.x * blockDim.x + threadIdx.x;
  if (i >= n4) return;
  float4 va = a[i], vb = b[i], vc = c[i], vd = d[i];
  float4 o;
  o.x = va.x + vb.x + vc.x + vd.x;
  o.y = va.y + vb.y + vc.y + vd.y;
  o.z = va.z + vb.z + vc.z + vd.z;
  o.w = va.w + vb.w + vc.w + vd.w;
  out[i] = o;
}

extern "C" void kernel_launch(void* const* d_in, const int* in_sizes, int n_in,
                              void* d_out, int out_size, void* d_ws, size_t ws_size,
                              hipStream_t stream) {
  const float* item_emb = (const float*)d_in[0];
  const float* weights  = (const float*)d_in[1];   // (3,128,128)
  const float* biases   = (const float*)d_in[2];   // (3,128)
  const int*   edge     = (const int*)d_in[3];     // (2,E) int32
  (void)d_in; (void)n_in; (void)out_size; (void)ws_size;

  const int n = GCN_N;
  const int E = in_sizes[3] / 2;
  const int embRows = in_sizes[0] / GCN_D;
  const float* x0src  = item_emb + (size_t)(embRows - n) * GCN_D;
  const int* srcIdx = edge;
  const int* dstIdx = edge + E;

  float* out = (float*)d_out;
  const size_t R = (size_t)n * GCN_D;
  float* oTotal = out;            // doubles as y scratch during layers
  float* oX0    = out + R;

  float* deg = (float*)d_ws;      // n floats; becomes dinv in place

  const int BT = 256;
  // degrees -> dinv
  k_deg_init <<<(n + BT - 1) / BT, BT, 0, stream>>>(deg, n);
  k_deg_edges<<<(E + BT - 1) / BT, BT, 0, stream>>>(dstIdx, deg, E);
  k_dinv     <<<(n + BT - 1) / BT, BT, 0, stream>>>(deg, n);

  // x0 copy into out region 1
  const int n4 = n * (GCN_D / 4);
  k_copy4<<<(n4 + BT - 1) / BT, BT, 0, stream>>>((const float4*)x0src, (float4*)oX0, n4);

  // 3 GCN layers
  const int tiles   = (n / 16) * (GCN_D / 16);     // 25000 waves
  const int gemmBlk = (tiles + 7) / 8;             // 8 waves per 256-thread block
  float* y = oTotal;
  for (int l = 0; l < 3; ++l) {
    const float* X  = out + (size_t)(1 + l) * R;
    float*       O  = out + (size_t)(2 + l) * R;
    const float* Wl = weights + (size_t)l * GCN_D * GCN_D;
    const float* bl = biases  + (size_t)l * GCN_D;

    k_gemm_wmma<<<gemmBlk, BT, 0, stream>>>(X, Wl, y, n);
    k_self_bias<<<(n4 + BT - 1) / BT, BT, 0, stream>>>(
        (const float4*)y, deg, (const float4*)bl, (float4*)O, n);
    const int eth = E * 32;
    k_scatter<<<(eth + BT - 1) / BT, BT, 0, stream>>>(
        (const float4*)y, srcIdx, dstIdx, deg, O, E);
  }

  // total = x0 + e1 + e2 + e3 (overwrites y scratch region)
  k_total<<<(n4 + BT - 1) / BT, BT, 0, stream>>>(
      (const float4*)(out + R), (const float4*)(out + 2 * R),
      (const float4*)(out + 3 * R), (const float4*)(out + 4 * R),
      (float4*)oTotal, n4);
}